// DetectionBEVLoss_60387240182386
// MI455X (gfx1250) — compile-verified
//
#include <hip/hip_runtime.h>
#include <hip/hip_bf16.h>
#include <math.h>

// ---------------------------------------------------------------------------
// DetectionBEVLoss for MI455X (gfx1250, wave32).
// Stage 1: grid-stride streaming kernel, per-pixel loss math, wave32 shfl
//          reduction + LDS cross-wave reduction -> 7 partials per block.
// Stage 2: single wave32; V_WMMA_F32_16X16X4_F32 with an all-ones A operand
//          as a 64-way f32 column-sum engine to fold the block partials,
//          then writes the 6 normalized loss scalars.
// ---------------------------------------------------------------------------

#define NCLS   10
#define F_ALPHA 0.25f
#define F_GAMMA 2.0f
#define F_BETA  1.0f
#define F_EPS   1e-07f
#define HW_     65536          // 256*256
#define NBLK    512            // stage-1 blocks (multiple of 64 for stage 2)
#define INV_PI2_4 0.40528473456935108f   // 4/pi^2

typedef __attribute__((ext_vector_type(2))) float v2f;
typedef __attribute__((ext_vector_type(8))) float v8f;

__device__ __forceinline__ float smooth_l1(float p, float t) {
    float d = fabsf(p - t);
    return (d < F_BETA) ? (0.5f * d * d / F_BETA) : (d - 0.5f * F_BETA);
}

struct Quad { float x[4], y[4]; };

__device__ __forceinline__ void make_corners(float cx, float cy, float dx, float dy,
                                             float yaw, Quad& q) {
    // base [[1,1],[-1,1],[-1,-1],[1,-1]] reordered [3,2,1,0]:
    const float mx[4] = { 1.f, -1.f, -1.f,  1.f };
    const float my[4] = {-1.f, -1.f,  1.f,  1.f };
    float hx = dx * 0.5f, hy = dy * 0.5f;
    float s, c;
    sincosf(yaw, &s, &c);
#pragma unroll
    for (int k = 0; k < 4; ++k) {
        float bx = mx[k] * hx, by = my[k] * hy;
        q.x[k] = bx * c - by * s + cx;
        q.y[k] = bx * s + by * c + cy;
    }
}

__device__ __forceinline__ float signed_area(const Quad& q) {
    float a = 0.f;
#pragma unroll
    for (int k = 0; k < 4; ++k) {
        int k1 = (k + 1) & 3;
        a += q.x[k] * q.y[k1] - q.y[k] * q.x[k1];
    }
    return 0.5f * a;
}

__device__ float rotated_iou(const Quad& A, const Quad& B) {
    float sA = signed_area(A), sB = signed_area(B);
    float sgnA = (sA > 0.f) ? 1.f : ((sA < 0.f) ? -1.f : 0.f);
    float sgnB = (sB > 0.f) ? 1.f : ((sB < 0.f) ? -1.f : 0.f);

    float vx[24], vy[24];
    int nv = 0;

    // vertices of A inside B (order matters for stable sort ties)
#pragma unroll
    for (int i = 0; i < 4; ++i) {
        bool in = true;
#pragma unroll
        for (int j = 0; j < 4; ++j) {
            int j1 = (j + 1) & 3;
            float ex = B.x[j1] - B.x[j], ey = B.y[j1] - B.y[j];
            float dx = A.x[i] - B.x[j], dy = A.y[i] - B.y[j];
            float cr = ex * dy - ey * dx;
            in = in && (cr * sgnB >= -1e-9f);
        }
        if (in) { vx[nv] = A.x[i]; vy[nv] = A.y[i]; ++nv; }
    }
    // vertices of B inside A
#pragma unroll
    for (int i = 0; i < 4; ++i) {
        bool in = true;
#pragma unroll
        for (int j = 0; j < 4; ++j) {
            int j1 = (j + 1) & 3;
            float ex = A.x[j1] - A.x[j], ey = A.y[j1] - A.y[j];
            float dx = B.x[i] - A.x[j], dy = B.y[i] - A.y[j];
            float cr = ex * dy - ey * dx;
            in = in && (cr * sgnA >= -1e-9f);
        }
        if (in) { vx[nv] = B.x[i]; vy[nv] = B.y[i]; ++nv; }
    }
    // edge-edge intersections (A-edge major, B-edge minor — matches reshape)
#pragma unroll
    for (int i = 0; i < 4; ++i) {
        int i1 = (i + 1) & 3;
        float dax = A.x[i1] - A.x[i], day = A.y[i1] - A.y[i];
#pragma unroll
        for (int j = 0; j < 4; ++j) {
            int j1 = (j + 1) & 3;
            float dbx = B.x[j1] - B.x[j], dby = B.y[j1] - B.y[j];
            float den = dax * dby - day * dbx;
            float aden = fabsf(den);
            float dsafe = (aden < 1e-12f) ? 1.f : den;
            float dpx = B.x[j] - A.x[i], dpy = B.y[j] - A.y[i];
            float t = (dpx * dby - dpy * dbx) / dsafe;
            float s = (dpx * day - dpy * dax) / dsafe;
            if ((aden > 1e-12f) && (t >= 0.f) && (t <= 1.f) && (s >= 0.f) && (s <= 1.f)) {
                vx[nv] = A.x[i] + t * dax;
                vy[nv] = A.y[i] + t * day;
                ++nv;
            }
        }
    }

    float inter = 0.f;
    if (nv >= 3) {
        float cx = 0.f, cy = 0.f;
        for (int k = 0; k < nv; ++k) { cx += vx[k]; cy += vy[k]; }
        float invn = 1.f / (float)nv;
        cx *= invn; cy *= invn;
        float ang[24];
        for (int k = 0; k < nv; ++k) ang[k] = atan2f(vy[k] - cy, vx[k] - cx);
        // stable insertion sort by angle (matches jnp stable argsort)
        for (int k = 1; k < nv; ++k) {
            float ka = ang[k], kx = vx[k], ky = vy[k];
            int j = k;
            while (j > 0 && ang[j - 1] > ka) {
                ang[j] = ang[j - 1]; vx[j] = vx[j - 1]; vy[j] = vy[j - 1];
                --j;
            }
            ang[j] = ka; vx[j] = kx; vy[j] = ky;
        }
        float acc = 0.f;
        for (int k = 0; k < nv; ++k) {
            int k1 = (k + 1 == nv) ? 0 : (k + 1);
            acc += vx[k] * vy[k1] - vy[k] * vx[k1];
        }
        inter = 0.5f * fabsf(acc);
    }

    float areaA = fabsf(sA), areaB = fabsf(sB);
    float uni = areaA + areaB - inter;
    return (uni > 1e-7f) ? (inter / uni) : 0.f;
}

// --------------------------- Stage 1 ---------------------------------------
__global__ __launch_bounds__(256)
void bev_loss_partial(const float* __restrict__ cls,
                      const float* __restrict__ regp,
                      const float* __restrict__ ioup,
                      const int*   __restrict__ ctg,
                      const float* __restrict__ regt,
                      const float* __restrict__ wgt,
                      const float* __restrict__ iout,
                      float* __restrict__ partials, int N) {
    float acc[7];
#pragma unroll
    for (int i = 0; i < 7; ++i) acc[i] = 0.f;

    for (int n = blockIdx.x * blockDim.x + threadIdx.x; n < N;
         n += gridDim.x * blockDim.x) {
        int b  = n >> 16;        // n / HW_
        int hw = n & 0xFFFF;     // n % HW_

        // ---- focal loss over 10 classes (coalesced strided loads) ----
        const float* cb = cls + (size_t)b * NCLS * HW_ + hw;
        float l[NCLS], mx = -1e30f;
#pragma unroll
        for (int c = 0; c < NCLS; ++c) { l[c] = cb[(size_t)c * HW_]; mx = fmaxf(mx, l[c]); }
        float ssum = 0.f;
#pragma unroll
        for (int c = 0; c < NCLS; ++c) ssum += expf(l[c] - mx);
        int t = ctg[n];
        int tl = min(max(t, 0), NCLS - 1);
        float pt = expf(l[tl] - mx) / ssum;
        pt = fminf(fmaxf(pt, F_EPS), 1.f - F_EPS);
        float at = (t > 0) ? F_ALPHA : (1.f - F_ALPHA);
        float om = 1.f - pt;
        float fl = -at * om * om * logf(pt);
        if (t >= 0) acc[0] += fl;

        // ---- regression branch only for positive anchors ----
        if (wgt[n] > 0.f) {
            const float* rp = regp + (size_t)b * 9 * HW_ + hw;
            const float* rt = regt + (size_t)b * 9 * HW_ + hw;
            float p[9], tv[9];
#pragma unroll
            for (int k = 0; k < 9; ++k) { p[k] = rp[(size_t)k * HW_]; tv[k] = rt[(size_t)k * HW_]; }

            Quad qp, qt;
            make_corners(p[0], p[1], p[3], p[4], p[6], qp);
            make_corners(tv[0], tv[1], tv[3], tv[4], tv[6], qt);
            float iou = rotated_iou(qp, qt);

            float ddx = p[0] - tv[0], ddy = p[1] - tv[1];
            float d2 = ddx * ddx + ddy * ddy;

            float mnx = qp.x[0], mxx = qp.x[0], mny = qp.y[0], mxy = qp.y[0];
#pragma unroll
            for (int k = 1; k < 4; ++k) {
                mnx = fminf(mnx, qp.x[k]); mxx = fmaxf(mxx, qp.x[k]);
                mny = fminf(mny, qp.y[k]); mxy = fmaxf(mxy, qp.y[k]);
            }
#pragma unroll
            for (int k = 0; k < 4; ++k) {
                mnx = fminf(mnx, qt.x[k]); mxx = fmaxf(mxx, qt.x[k]);
                mny = fminf(mny, qt.y[k]); mxy = fmaxf(mxy, qt.y[k]);
            }
            float ex = mxx - mnx, ey = mxy - mny;
            float c2 = fmaxf(ex * ex + ey * ey, F_EPS);

            float atp = atanf(p[4]  / fmaxf(p[3],  F_EPS));
            float att = atanf(tv[4] / fmaxf(tv[3], F_EPS));
            float dv = atp - att;
            float vv = INV_PI2_4 * dv * dv;
            float ac = vv / (1.f - iou + vv + F_EPS);

            acc[1] += 1.f - iou + d2 / c2 + ac * vv;
            acc[2] += smooth_l1(p[2], tv[2]);
            acc[3] += smooth_l1(p[5], tv[5]);
            acc[4] += smooth_l1(p[7], tv[7]) + smooth_l1(p[8], tv[8]);

            float x = ioup[n], y = iout[n];
            acc[5] += fmaxf(x, 0.f) - x * y + log1pf(expf(-fabsf(x)));
            acc[6] += 1.f;   // num_pos
        }
    }

    // ---- wave32 tree reduction, then cross-wave LDS reduction ----
#pragma unroll
    for (int i = 0; i < 7; ++i)
#pragma unroll
        for (int off = 16; off > 0; off >>= 1)
            acc[i] += __shfl_down(acc[i], off, 32);

    __shared__ float red[8][7];
    int lane = threadIdx.x & 31, wid = threadIdx.x >> 5;
    if (lane == 0)
#pragma unroll
        for (int i = 0; i < 7; ++i) red[wid][i] = acc[i];
    __syncthreads();

    if (threadIdx.x == 0) {
#pragma unroll
        for (int i = 0; i < 7; ++i) {
            float s = 0.f;
#pragma unroll
            for (int wv = 0; wv < 8; ++wv) s += red[wv][i];
            partials[(size_t)blockIdx.x * 7 + i] = s;
        }
    }
}

// --------------------------- Stage 2 (WMMA reduction) ----------------------
// One wave32. For each accumulator a: fold NBLK partials 64-at-a-time with
// V_WMMA_F32_16X16X4_F32, A = ones(16x4), B = data, C = running D.
// D ends as 16 identical rows of column sums -> sum(all D)/16 = grand total.
__global__ __launch_bounds__(32)
void bev_loss_final(const float* __restrict__ partials, float* __restrict__ out) {
    int lane = threadIdx.x;
    float tot[7];

    v2f ones; ones.x = 1.f; ones.y = 1.f;

#pragma unroll
    for (int a = 0; a < 7; ++a) {
        v8f d = {};   // zero accumulator
        for (int base = 0; base < NBLK; base += 64) {
            int e0 = base + lane * 2;
            v2f bv;
            bv.x = partials[(size_t)e0 * 7 + a];
            bv.y = partials[(size_t)(e0 + 1) * 7 + a];
            // (neg_a, A, neg_b, B, c_mod, C, reuse_a, reuse_b)
            d = __builtin_amdgcn_wmma_f32_16x16x4_f32(
                    false, ones, false, bv, (short)0, d, false, false);
        }
        float s = d[0] + d[1] + d[2] + d[3] + d[4] + d[5] + d[6] + d[7];
#pragma unroll
        for (int off = 16; off > 0; off >>= 1) s += __shfl_down(s, off, 32);
        tot[a] = s * (1.0f / 16.0f);   // valid at lane 0
    }

    if (lane == 0) {
        float np = fmaxf(tot[6], 1.f);
        float inv = 1.f / np;
#pragma unroll
        for (int i = 0; i < 6; ++i) out[i] = tot[i] * inv;
    }
}

// --------------------------- Host launcher ---------------------------------
extern "C" void kernel_launch(void* const* d_in, const int* in_sizes, int n_in,
                              void* d_out, int out_size, void* d_ws, size_t ws_size,
                              hipStream_t stream) {
    const float* cls  = (const float*)d_in[0];
    const float* regp = (const float*)d_in[1];
    const float* ioup = (const float*)d_in[2];
    const int*   ctg  = (const int*)  d_in[3];
    const float* regt = (const float*)d_in[4];
    const float* wgt  = (const float*)d_in[5];
    const float* iout = (const float*)d_in[6];
    float* out      = (float*)d_out;
    float* partials = (float*)d_ws;   // NBLK*7 floats = 14 KB

    const int N = in_sizes[3];        // B*H*W = 524288

    bev_loss_partial<<<NBLK, 256, 0, stream>>>(cls, regp, ioup, ctg, regt, wgt,
                                               iout, partials, N);
    bev_loss_final<<<1, 32, 0, stream>>>(partials, out);
}